// SoftSkeletonize_20358144983651
// MI455X (gfx1250) — compile-verified
//
#include <hip/hip_runtime.h>
#include <stdint.h>

// Fixed problem shape from the reference: (N=2, C=1, D=192, H=256, W=256)
constexpr int N_ = 2;
constexpr int D_ = 192;
constexpr int H_ = 256;
constexpr int W_ = 256;
constexpr int HW_ = H_ * W_;          // 65536 (power of two -> shifts)

// Output tile per 256-thread block
constexpr int OX = 32;
constexpr int OY = 8;
constexpr int OZ = 8;
// Input tile with halo 2 on each side
constexpr int IX = 36;
constexpr int IY = 12;
constexpr int IZ = 12;
constexpr int ZT = D_ / OZ;           // 24 z-tiles per volume

__device__ __forceinline__ float min3f(float a, float b, float c) {
    return fminf(fminf(a, b), c);
}
__device__ __forceinline__ float max3f(float a, float b, float c) {
    return fmaxf(fmaxf(a, b), c);
}

__global__ __launch_bounds__(256) void soft_open_relu_kernel(
    const float* __restrict__ img, float* __restrict__ out)
{
    __shared__ float s_in[IZ * IY * IX];   // 5184 : input tile (halo 2), alive whole kernel
    __shared__ float s_a [12 * 12 * 34];   // 4896 : min-x pass, later reused for max-x
    __shared__ float s_b [12 * 10 * 34];   // 4080 : min-xy pass, later reused for max-xy
    __shared__ float s_er[10 * 10 * 34];   // 3400 : eroded field with halo 1

    const int tid = threadIdx.x;
    const int x0  = blockIdx.x * OX;
    const int y0  = blockIdx.y * OY;
    const int zb  = blockIdx.z;
    const int n   = zb / ZT;               // div by constant -> mul/shift
    const int z0  = (zb % ZT) * OZ;

    const size_t nbase = (size_t)n * (D_ * HW_);
    const float* base = img + nbase;       // block-uniform 64-bit base -> SGPR pair (GVS mode)

    // ---- Stage 0: async DMA of the edge-clamped halo tile into LDS -------------
    // global_load_async_to_lds_b32: VDST = per-lane LDS byte address,
    // VADDR = per-lane 32-bit byte offset, SADDR = uniform 64-bit base.
    // Tracked by ASYNCcnt, drained with s_wait_asynccnt.
    for (int i = tid; i < IZ * IY * IX; i += 256) {
        int x = i % IX;
        int t = i / IX;
        int y = t % IY;
        int z = t / IY;
        int gx = x0 - 2 + x; gx = gx < 0 ? 0 : (gx >= W_ ? W_ - 1 : gx);
        int gy = y0 - 2 + y; gy = gy < 0 ? 0 : (gy >= H_ ? H_ - 1 : gy);
        int gz = z0 - 2 + z; gz = gz < 0 ? 0 : (gz >= D_ ? D_ - 1 : gz);
        unsigned goff  = (unsigned)((gz * HW_ + gy * W_ + gx) * 4);
        unsigned laddr = (unsigned)(uintptr_t)(&s_in[i]);  // low 32 bits = LDS offset
        asm volatile("global_load_async_to_lds_b32 %0, %1, %2"
                     :: "v"(laddr), "v"(goff), "s"(base)
                     : "memory");
    }
    asm volatile("s_wait_asynccnt 0" ::: "memory");
    __syncthreads();

    // ---- Pass 1: erode, min over x (3-tap) -> s_a[12z][12y][34x] ---------------
    for (int i = tid; i < 12 * 12 * 34; i += 256) {
        int x = i % 34;
        int t = i / 34;
        int y = t % 12;
        int z = t / 12;
        const float* p = &s_in[z * (IY * IX) + y * IX + x];
        s_a[i] = min3f(p[0], p[1], p[2]);
    }
    __syncthreads();

    // ---- Pass 2: erode, min over y -> s_b[12z][10y][34x] -----------------------
    for (int i = tid; i < 12 * 10 * 34; i += 256) {
        int x = i % 34;
        int t = i / 34;
        int y = t % 10;
        int z = t / 10;
        const float* p = &s_a[z * (12 * 34) + y * 34 + x];
        s_b[i] = min3f(p[0], p[34], p[68]);
    }
    __syncthreads();

    // ---- Pass 3: erode, min over z -> s_er[10z][10y][34x]; -inf outside volume -
    // (the following dilate must ignore eroded samples that lie outside the volume)
    for (int i = tid; i < 10 * 10 * 34; i += 256) {
        int x = i % 34;
        int t = i / 34;
        int y = t % 10;
        int z = t / 10;
        const float* p = &s_b[z * (10 * 34) + y * 34 + x];
        float v = min3f(p[0], p[340], p[680]);
        int gx = x0 - 1 + x;
        int gy = y0 - 1 + y;
        int gz = z0 - 1 + z;
        bool oob = ((unsigned)gx >= (unsigned)W_) |
                   ((unsigned)gy >= (unsigned)H_) |
                   ((unsigned)gz >= (unsigned)D_);
        s_er[i] = oob ? -__builtin_huge_valf() : v;
    }
    __syncthreads();

    // ---- Pass 4: dilate, max over x -> reuse s_a[10z][10y][32x] ----------------
    for (int i = tid; i < 10 * 10 * 32; i += 256) {
        int x = i % 32;
        int t = i / 32;
        int y = t % 10;
        int z = t / 10;
        const float* p = &s_er[z * 340 + y * 34 + x];
        s_a[i] = max3f(p[0], p[1], p[2]);
    }
    __syncthreads();

    // ---- Pass 5: dilate, max over y -> reuse s_b[10z][8y][32x] -----------------
    for (int i = tid; i < 10 * 8 * 32; i += 256) {
        int x = i % 32;
        int t = i / 32;
        int y = t % 8;
        int z = t / 8;
        const float* p = &s_a[z * 320 + y * 32 + x];
        s_b[i] = max3f(p[0], p[32], p[64]);
    }
    __syncthreads();

    // ---- Pass 6: dilate max over z, residual ReLU, coalesced store -------------
    for (int i = tid; i < OZ * OY * OX; i += 256) {
        int x = i % 32;
        int t = i / 32;
        int y = t % 8;
        int z = t / 8;
        const float* p = &s_b[z * 256 + y * 32 + x];
        float opened = max3f(p[0], p[256], p[512]);
        float orig   = s_in[(z + 2) * (IY * IX) + (y + 2) * IX + (x + 2)];
        float r = fmaxf(orig - opened, 0.0f);
        out[nbase + (size_t)((z0 + z) * HW_ + (y0 + y) * W_ + (x0 + x))] = r;
    }
}

extern "C" void kernel_launch(void* const* d_in, const int* in_sizes, int n_in,
                              void* d_out, int out_size, void* d_ws, size_t ws_size,
                              hipStream_t stream) {
    (void)in_sizes; (void)n_in; (void)out_size; (void)d_ws; (void)ws_size;
    const float* img = (const float*)d_in[0];
    float* out = (float*)d_out;
    dim3 block(256, 1, 1);
    dim3 grid(W_ / OX, H_ / OY, ZT * N_);
    soft_open_relu_kernel<<<grid, block, 0, stream>>>(img, out);
}